// VectorQuantizer_81432579932437
// MI455X (gfx1250) — compile-verified
//
#include <hip/hip_runtime.h>
#include <math.h>

typedef float v2f __attribute__((ext_vector_type(2)));
typedef float v8f __attribute__((ext_vector_type(8)));

#define VQ_D 64
#define VQ_K 512
#define VQ_N (64 * 64 * 64)   // 262144 input vectors
#define LDSS 66               // padded LDS row stride in floats (bank-conflict free)
#define ENORM_OFF (VQ_K * LDSS)  // float offset of ||e||^2 table in LDS

// ---------------------------------------------------------------------------
// Prep: ||e_n||^2 for each codebook entry
// ---------------------------------------------------------------------------
__global__ __launch_bounds__(512)
void vq_prep_kernel(const float* __restrict__ E, float* __restrict__ enorm) {
  int n = blockIdx.x * blockDim.x + threadIdx.x;
  if (n < VQ_K) {
    const float* row = E + (size_t)n * VQ_D;
    float s = 0.f;
#pragma unroll
    for (int d = 0; d < VQ_D; ++d) { float v = row[d]; s += v * v; }
    enorm[n] = s;
  }
}

// ---------------------------------------------------------------------------
// Main: distance GEMM via V_WMMA_F32_16X16X4_F32 (two accumulator chains),
// argmin, gather, stats. Block = 256 threads = 8 waves; 128 rows per block.
// ---------------------------------------------------------------------------
__global__ __launch_bounds__(256)
void vq_main_kernel(const float* __restrict__ X, const float* __restrict__ E,
                    const float* __restrict__ enorm, float* __restrict__ OUT,
                    unsigned int* __restrict__ counts, float* __restrict__ errAcc) {
  extern __shared__ float eLds[];   // codebook (padded) + ||e||^2 table
  const int tid = threadIdx.x;

  // ---- stage full codebook into padded LDS (float2 granularity, 8B aligned)
  for (int i = tid; i < VQ_K * VQ_D / 2; i += 256) {
    int n = i >> 5;               // 32 float2 per 64-float row
    int d = (i & 31) << 1;
    float2 v = *(const float2*)(E + (size_t)n * VQ_D + d);
    *(float2*)(eLds + n * LDSS + d) = v;
  }
  // ---- stage ||e||^2 table
  for (int i = tid; i < VQ_K; i += 256) eLds[ENORM_OFF + i] = enorm[i];
  __syncthreads();

  const int lane  = tid & 31;
  const int wave  = tid >> 5;
  const int mrow  = lane & 15;
  const int khalf = lane >> 4;    // 0: k-pair {0,1}; 1: k-pair {2,3}
  const size_t row0 = (size_t)blockIdx.x * 128 + (size_t)wave * 16;

  // ---- A fragment: 16x64 panel, ISA 16x4-f32 layout (raw x values) ----
  const float* xrow = X + (row0 + (size_t)mrow) * VQ_D;
  float a[32];
  float xnorm = 0.f;
#pragma unroll
  for (int s = 0; s < 16; ++s) {
    float2 v = *(const float2*)(xrow + 4 * s + 2 * khalf);
    xnorm += v.x * v.x + v.y * v.y;
    a[2 * s]     = v.x;
    a[2 * s + 1] = v.y;
  }
  xnorm += __shfl_xor(xnorm, 16);   // halves cover disjoint column sets

  float minval[8];
  int   minidx[8];
#pragma unroll
  for (int v = 0; v < 8; ++v) { minval[v] = 3.4e38f; minidx[v] = 0; }

  // ---- sweep 512 codes, two 16-column tiles per iteration (dual chains) ----
  for (int tt = 0; tt < VQ_K / 32; ++tt) {
    const int n0 = tt * 32 + mrow;        // C/B layout: N = lane & 15
    const int n1 = n0 + 16;
    v8f c0 = {};                          // accumulate raw x.e (C starts at 0)
    v8f c1 = {};
    const float* b0 = eLds + n0 * LDSS + 2 * khalf;
    const float* b1 = eLds + n1 * LDSS + 2 * khalf;
#pragma unroll
    for (int s = 0; s < 16; ++s) {        // K = 64 in steps of 4
      v2f av; av.x = a[2 * s]; av.y = a[2 * s + 1];
      float2 l0 = *(const float2*)(b0 + 4 * s);
      float2 l1 = *(const float2*)(b1 + 4 * s);
      v2f bv0; bv0.x = l0.x; bv0.y = l0.y;
      v2f bv1; bv1.x = l1.x; bv1.y = l1.y;
      c0 = __builtin_amdgcn_wmma_f32_16x16x4_f32(false, av, false, bv0,
                                                 (short)0, c0, false, false);
      c1 = __builtin_amdgcn_wmma_f32_16x16x4_f32(false, av, false, bv1,
                                                 (short)0, c1, false, false);
    }
    const float en0 = eLds[ENORM_OFF + n0];
    const float en1 = eLds[ENORM_OFF + n1];
#pragma unroll
    for (int v = 0; v < 8; ++v) {
      float s0 = fmaf(-2.f, c0[v], en0);  // ||e||^2 - 2 x.e  (row m = v+8*khalf)
      if (s0 < minval[v]) { minval[v] = s0; minidx[v] = n0; }
      float s1 = fmaf(-2.f, c1[v], en1);
      if (s1 < minval[v]) { minval[v] = s1; minidx[v] = n1; }
    }
  }

  // ---- argmin butterfly across each 16-lane half (first-min tie-break) ----
  float errsum = 0.f;
  int winIdx[8];
#pragma unroll
  for (int v = 0; v < 8; ++v) {
    float bv = minval[v];
    int   bi = minidx[v];
#pragma unroll
    for (int m = 8; m >= 1; m >>= 1) {
      float ov = __shfl_xor(bv, m, 16);
      int   oi = __shfl_xor(bi, m, 16);
      if (ov < bv || (ov == bv && oi < bi)) { bv = ov; bi = oi; }
    }
    float xn = __shfl(xnorm, v + 8 * khalf);   // ||x||^2 of this group's row
    errsum += xn + bv;                          // ||x - e_win||^2
    winIdx[v] = bi;
  }
  if ((lane & 15) == 0) {                       // lane0: rows v; lane16: rows v+8
    atomicAdd(errAcc, errsum);
#pragma unroll
    for (int v = 0; v < 8; ++v) atomicAdd(&counts[winIdx[v]], 1u);
  }

  // ---- gather winning codes from LDS, write quantized_st ----
#pragma unroll
  for (int v = 0; v < 8; ++v) {
    int i0 = __shfl(winIdx[v], 0);
    int i1 = __shfl(winIdx[v], 16);
    float2 q0 = *(const float2*)(eLds + i0 * LDSS + 2 * lane);
    float2 q1 = *(const float2*)(eLds + i1 * LDSS + 2 * lane);
    *(float2*)(OUT + (row0 + v) * VQ_D + 2 * lane)     = q0;
    *(float2*)(OUT + (row0 + v + 8) * VQ_D + 2 * lane) = q1;
  }
}

// ---------------------------------------------------------------------------
// Final: loss + perplexity scalars
// ---------------------------------------------------------------------------
__global__ __launch_bounds__(512)
void vq_final_kernel(const unsigned int* __restrict__ counts,
                     const float* __restrict__ errAcc,
                     float* __restrict__ tail) {
  __shared__ float red[512];
  int n = threadIdx.x;
  float p = (float)counts[n] * (1.0f / (float)VQ_N);
  red[n] = p * logf(p + 1e-10f);
  __syncthreads();
  for (int s = 256; s > 0; s >>= 1) {
    if (n < s) red[n] += red[n + s];
    __syncthreads();
  }
  if (n == 0) {
    tail[0] = 1.25f * errAcc[0] * (1.0f / ((float)VQ_N * (float)VQ_D));
    tail[1] = expf(-red[0]);
  }
}

// ---------------------------------------------------------------------------
extern "C" void kernel_launch(void* const* d_in, const int* in_sizes, int n_in,
                              void* d_out, int out_size, void* d_ws, size_t ws_size,
                              hipStream_t stream) {
  const float* X = (const float*)d_in[0];   // inputs  [64,64,64,64] f32
  const float* E = (const float*)d_in[1];   // codebook [512,64] f32
  float* OUT = (float*)d_out;               // quantized (16777216) + loss + perplexity

  float*        enorm  = (float*)d_ws;                        // 512 f32
  unsigned int* counts = (unsigned int*)((char*)d_ws + 2048); // 512 u32
  float*        errAcc = (float*)((char*)d_ws + 4096);        // 1 f32

  // zero counters + error accumulator (capturable stream memset)
  hipMemsetAsync((char*)d_ws + 2048, 0, 2048 + 4, stream);

  vq_prep_kernel<<<2, 256, 0, stream>>>(E, enorm);

  const size_t ldsBytes = (size_t)(VQ_K * LDSS + VQ_K) * sizeof(float); // 134 KB
  vq_main_kernel<<<VQ_N / 128, 256, ldsBytes, stream>>>(X, E, enorm, OUT,
                                                        counts, errAcc);

  vq_final_kernel<<<1, 512, 0, stream>>>(counts, errAcc, OUT + (out_size - 2));
}